// LIFSpike_1580547965313
// MI455X (gfx1250) — compile-verified
//
#include <hip/hip_runtime.h>

// LIF forward over T=8 for x:[B=32,T=8,C=128,H=32,W=32] fp32.
// Pure streaming recurrence: HBM-bandwidth bound (256 MiB @ 23.3 TB/s ~ 11.5us).
// Strategy: one thread per float4 along W per (b,c,h); all 8 time slices loaded
// up-front as nontemporal B128 loads (8-deep MLP), recurrence in registers,
// nontemporal B128 stores. No matrix structure -> WMMA intentionally unused.

typedef float v4f __attribute__((ext_vector_type(4)));

#define LIF_TAU    0.5f
#define LIF_THRESH 1.0f

// Compile-time shape constants (match the reference harness shapes).
#define LIF_B     32
#define LIF_T     8
#define LIF_CHW   (128 * 32 * 32)       // 131072 floats per (b,t) slice
#define LIF_CHW4  (LIF_CHW / 4)         // 32768 float4 per (b,t) slice (2^15)
#define LIF_N     (LIF_B * LIF_CHW4)    // 1048576 thread work items

__global__ __launch_bounds__(256) void lif_forward_kernel(
    const v4f* __restrict__ x4, v4f* __restrict__ out4, int n)
{
    int idx = blockIdx.x * blockDim.x + threadIdx.x;
    if (idx >= n) return;

    // idx = b * CHW4 + r ; CHW4 is a power of two -> shift/mask
    int b = idx >> 15;                 // idx / LIF_CHW4
    int r = idx & (LIF_CHW4 - 1);      // idx % LIF_CHW4

    // element (b, t, r) lives at b*T*CHW4 + t*CHW4 + r (in float4 units)
    size_t base = (size_t)b * (LIF_T * LIF_CHW4) + (size_t)r;

    // Issue all 8 independent time-slice loads up front: 8 outstanding
    // global_load_b128 (nontemporal) per thread -> deep MLP, single wait.
    v4f xt[LIF_T];
#pragma unroll
    for (int t = 0; t < LIF_T; ++t) {
        xt[t] = __builtin_nontemporal_load(&x4[base + (size_t)t * LIF_CHW4]);
    }

    // Serial LIF recurrence entirely in VGPRs.
    v4f mem = (v4f)(0.0f);
#pragma unroll
    for (int t = 0; t < LIF_T; ++t) {
        v4f s;
#pragma unroll
        for (int k = 0; k < 4; ++k) {
            float m = mem[k] * LIF_TAU + xt[t][k];   // v_fma
            bool fire = (m > LIF_THRESH);            // v_cmp_gt_f32
            s[k]   = fire ? 1.0f : 0.0f;             // v_cndmask
            mem[k] = fire ? 0.0f : m;                // v_cndmask (hard reset)
        }
        __builtin_nontemporal_store(s, &out4[base + (size_t)t * LIF_CHW4]);
    }
}

extern "C" void kernel_launch(void* const* d_in, const int* in_sizes, int n_in,
                              void* d_out, int out_size, void* d_ws, size_t ws_size,
                              hipStream_t stream)
{
    (void)n_in; (void)out_size; (void)d_ws; (void)ws_size;

    const v4f* x4   = (const v4f*)d_in[0];
    v4f*       out4 = (v4f*)d_out;

    // in_sizes[0] = B*T*C*H*W ; each thread handles T steps of one float4.
    int n = in_sizes[0] / (LIF_T * 4);   // = LIF_N = 1048576

    const int block = 256;               // 8 wave32 per block
    const int grid  = (n + block - 1) / block;   // 4096
    lif_forward_kernel<<<grid, block, 0, stream>>>(x4, out4, n);
}